// Expert_25555055411706
// MI455X (gfx1250) — compile-verified
//
#include <hip/hip_runtime.h>
#include <hip/hip_bf16.h>

// Problem constants (match reference)
#define BS      8
#define LSEQ    2048
#define DM      512
#define DI      1024
#define DSTATE  16
#define DCONV   4
#define DTR     32
#define NEXP    4
#define NLAYER  2

typedef __attribute__((ext_vector_type(2))) float v2f;
typedef __attribute__((ext_vector_type(8))) float v8f;

#define MT   128  // M block tile
#define NT   64   // N block tile
#define KB   64   // K chunk staged in LDS
#define LDP  68   // padded LDS row stride (floats): bank stride 4, conflict-free frag reads

__device__ __forceinline__ float silu_f(float x) {
    return x / (1.0f + __expf(-x));
}
__device__ __forceinline__ float softplus_f(float x) {
    return (x > 20.0f) ? x : log1pf(__expf(x));
}
__device__ __forceinline__ int expert_of(const int* eid, int b) {
    int e = eid[b] - 1;
    return (e < 0) ? 0 : ((e > NEXP - 1) ? NEXP - 1 : e);
}

// -----------------------------------------------------------------------------
// fp32 WMMA GEMM: C[b] (M x N) = A[b] (M x K, lda) @ W[e,l] (K x N, ldb)
// 128x64 block tile, 8 waves; each wave owns a 2x2 grid of 16x16 subtiles
// (2 A-frag + 2 B-frag ds_load_b64 per 4 WMMAs). B tile stored transposed in
// LDS so every fragment is a contiguous 8-byte pair (no v_mov assembly).
// mode 0: plain store. mode 1: softplus(acc + bias[e,l][col]) (dt projection).
// -----------------------------------------------------------------------------
__global__ __launch_bounds__(256)
void gemm_wmma_f32(const float* __restrict__ A,
                   const float* __restrict__ W,
                   float* __restrict__ C,
                   const int* __restrict__ eid,
                   int layer, long long wstride,
                   int M, int N, int K,
                   int lda, int ldb, int ldc,
                   const float* __restrict__ bias, int mode) {
    __shared__ float As[MT * LDP];   // [m][k]
    __shared__ float Bs[NT * LDP];   // [n][k]  (transposed)

    const int b  = blockIdx.z;
    const int e  = expert_of(eid, b);
    const float* Ab = A + (size_t)b * (size_t)M * (size_t)lda;
    const float* Wb = W + (size_t)(e * NLAYER + layer) * (size_t)wstride;
    float*       Cb = C + (size_t)b * (size_t)M * (size_t)ldc;

    const int n0 = blockIdx.x * NT;
    const int m0 = blockIdx.y * MT;

    const int tid  = threadIdx.x;
    const int wave = tid >> 5;
    const int lane = tid & 31;
    const int wm   = wave & 3;        // M subtiles wm and wm+4   (of 8)
    const int wn   = wave >> 2;       // N subtiles wn and wn+2   (of 4)
    const int half = lane >> 4;       // lane half selects K pair (ISA A/B layout)
    const int lm   = lane & 15;

    v8f acc00 = {}, acc01 = {}, acc10 = {}, acc11 = {};

    const int arow0 = (wm * 16 + lm) * LDP + half * 2;
    const int arow1 = ((wm + 4) * 16 + lm) * LDP + half * 2;
    const int brow0 = (wn * 16 + lm) * LDP + half * 2;
    const int brow1 = ((wn + 2) * 16 + lm) * LDP + half * 2;

    for (int k0 = 0; k0 < K; k0 += KB) {
        // ---- stage A (MT x KB) row-major ----------------------------------
        #pragma unroll
        for (int i = 0; i < (MT * KB) / (256 * 4); ++i) {   // 8 float4 per thread
            int j = tid + i * 256;
            int r = j >> 4;               // 0..127 (m)
            int c = (j & 15) << 2;        // 0..60  (k)
            float4 va = make_float4(0.f, 0.f, 0.f, 0.f);
            if (k0 + c < K)
                va = *(const float4*)(Ab + (size_t)(m0 + r) * lda + (k0 + c));
            *(float4*)(&As[r * LDP + c]) = va;
        }
        // ---- stage B (KB x NT) transposed into Bs[n][k] -------------------
        #pragma unroll
        for (int i = 0; i < (NT * KB) / (256 * 4); ++i) {   // 4 float4 per thread
            int j = tid + i * 256;
            int k = j >> 4;               // 0..63 (k)
            int n = (j & 15) << 2;        // 0..60 (n)
            float4 vb = make_float4(0.f, 0.f, 0.f, 0.f);
            if (k0 + k < K)
                vb = *(const float4*)(Wb + (size_t)(k0 + k) * ldb + (n0 + n));
            Bs[(n + 0) * LDP + k] = vb.x;
            Bs[(n + 1) * LDP + k] = vb.y;
            Bs[(n + 2) * LDP + k] = vb.z;
            Bs[(n + 3) * LDP + k] = vb.w;
        }
        // prefetch next weight chunk (global_prefetch_b8)
        if (k0 + KB < K)
            __builtin_prefetch(Wb + (size_t)(k0 + KB + (tid >> 2)) * ldb + n0 + ((tid & 3) << 4), 0, 0);
        __syncthreads();

        // ---- 16 WMMA steps over the KB chunk ------------------------------
        #pragma unroll
        for (int kk = 0; kk < KB; kk += 4) {
            const v2f a0 = *(const v2f*)(&As[arow0 + kk]);
            const v2f a1 = *(const v2f*)(&As[arow1 + kk]);
            const v2f b0 = *(const v2f*)(&Bs[brow0 + kk]);
            const v2f b1 = *(const v2f*)(&Bs[brow1 + kk]);
            acc00 = __builtin_amdgcn_wmma_f32_16x16x4_f32(false, a0, false, b0,
                                                          (short)0, acc00, false, false);
            acc01 = __builtin_amdgcn_wmma_f32_16x16x4_f32(false, a0, false, b1,
                                                          (short)0, acc01, false, false);
            acc10 = __builtin_amdgcn_wmma_f32_16x16x4_f32(false, a1, false, b0,
                                                          (short)0, acc10, false, false);
            acc11 = __builtin_amdgcn_wmma_f32_16x16x4_f32(false, a1, false, b1,
                                                          (short)0, acc11, false, false);
        }
        __syncthreads();
    }

    // ---- epilogue: C/D layout = VGPR r: lanes 0-15 M=r, lanes 16-31 M=r+8 --
    const float* bb = (mode == 1) ? (bias + (size_t)(e * NLAYER + layer) * (size_t)N)
                                  : (const float*)nullptr;
    const int col0 = n0 + wn * 16 + lm;
    const int col1 = n0 + (wn + 2) * 16 + lm;
    #pragma unroll
    for (int r = 0; r < 8; ++r) {
        const int row0 = m0 + wm * 16 + half * 8 + r;
        const int row1 = m0 + (wm + 4) * 16 + half * 8 + r;
        float v00 = acc00[r], v01 = acc01[r];
        float v10 = acc10[r], v11 = acc11[r];
        if (mode == 1) {
            const float bc0 = bb[col0], bc1 = bb[col1];
            v00 = softplus_f(v00 + bc0);
            v01 = softplus_f(v01 + bc1);
            v10 = softplus_f(v10 + bc0);
            v11 = softplus_f(v11 + bc1);
        }
        Cb[(size_t)row0 * ldc + col0] = v00;
        Cb[(size_t)row0 * ldc + col1] = v01;
        Cb[(size_t)row1 * ldc + col0] = v10;
        Cb[(size_t)row1 * ldc + col1] = v11;
    }
}

// -----------------------------------------------------------------------------
// Depthwise causal conv (k=4) + bias + SiLU over the xc half of xr.
// -----------------------------------------------------------------------------
__global__ __launch_bounds__(256)
void conv_silu_kernel(const float* __restrict__ xr,
                      const float* __restrict__ conv_w,
                      const float* __restrict__ conv_b,
                      const int* __restrict__ eid,
                      int layer,
                      float* __restrict__ xc) {
    const size_t idx = (size_t)blockIdx.x * 256 + threadIdx.x;   // B*L*DI
    const int d = (int)(idx % DI);
    const int t = (int)((idx / DI) % LSEQ);
    const int b = (int)(idx / ((size_t)LSEQ * DI));
    const int e = expert_of(eid, b);

    const float* w = conv_w + ((size_t)(e * NLAYER + layer) * DI + d) * DCONV;
    float acc = conv_b[(size_t)(e * NLAYER + layer) * DI + d];
    const float* col = xr + (size_t)b * LSEQ * (2 * DI) + d;   // xc half: cols [0,DI)
    #pragma unroll
    for (int k = 0; k < DCONV; ++k) {
        const int tt = t + k - (DCONV - 1);
        if (tt >= 0) acc = fmaf(col[(size_t)tt * (2 * DI)], w[k], acc);
    }
    xc[idx] = silu_f(acc);
}

// -----------------------------------------------------------------------------
// Selective scan: one thread per channel; h[16] and A[16] in registers.
// Fuses y = (h@C + D*xc) * silu(res). y may alias dt (read-before-write per t).
// -----------------------------------------------------------------------------
__global__ __launch_bounds__(256)
void scan_kernel(const float* __restrict__ proj,
                 const float* dt_in,
                 const float* __restrict__ xc,
                 const float* __restrict__ xr,
                 const float* __restrict__ A_log,
                 const float* __restrict__ Dvec,
                 const int* __restrict__ eid,
                 int layer,
                 float* y_out) {
    const int b = blockIdx.y;
    const int d = blockIdx.x * 256 + threadIdx.x;       // channel
    const int e = expert_of(eid, b);

    const float* al = A_log + ((size_t)(e * NLAYER + layer) * DI + d) * DSTATE;
    float A[DSTATE], h[DSTATE];
    #pragma unroll
    for (int s = 0; s < DSTATE; ++s) { A[s] = -__expf(al[s]); h[s] = 0.0f; }
    const float Dp = Dvec[(size_t)(e * NLAYER + layer) * DI + d];

    const float* pb  = proj  + (size_t)b * LSEQ * 64;
    const float* db  = dt_in + (size_t)b * LSEQ * DI;
    const float* xb  = xc    + (size_t)b * LSEQ * DI;
    const float* rb  = xr    + (size_t)b * LSEQ * (2 * DI) + DI;   // res half
    float*       yb  = y_out + (size_t)b * LSEQ * DI;

    for (int t = 0; t < LSEQ; ++t) {
        const float dtv = db[(size_t)t * DI + d];
        const float xv  = xb[(size_t)t * DI + d];
        const float dx  = dtv * xv;
        const float* pr = pb + (size_t)t * 64;
        float acc = 0.0f;
        #pragma unroll
        for (int s = 0; s < DSTATE; ++s) {
            const float Bt = pr[DTR + s];              // lane-uniform broadcast
            const float Ct = pr[DTR + DSTATE + s];
            h[s] = fmaf(__expf(dtv * A[s]), h[s], dx * Bt);
            acc  = fmaf(h[s], Ct, acc);
        }
        const float r = rb[(size_t)t * (2 * DI) + d];
        yb[(size_t)t * DI + d] = fmaf(Dp, xv, acc) * silu_f(r);
    }
}

// -----------------------------------------------------------------------------
// Passthrough for expert_id == 0 samples.
// -----------------------------------------------------------------------------
__global__ __launch_bounds__(256)
void final_select_kernel(const float* __restrict__ x,
                         const int* __restrict__ eid,
                         float* __restrict__ out) {
    const size_t idx = (size_t)blockIdx.x * 256 + threadIdx.x;   // B*L*DM
    const int b = (int)(idx / ((size_t)LSEQ * DM));
    if (eid[b] == 0) out[idx] = x[idx];
}

extern "C" void kernel_launch(void* const* d_in, const int* in_sizes, int n_in,
                              void* d_out, int out_size, void* d_ws, size_t ws_size,
                              hipStream_t stream) {
    const float* x      = (const float*)d_in[0];
    const int*   eid    = (const int*)  d_in[1];
    const float* W_in   = (const float*)d_in[2];
    const float* conv_w = (const float*)d_in[3];
    const float* conv_b = (const float*)d_in[4];
    const float* W_x    = (const float*)d_in[5];
    const float* W_dt   = (const float*)d_in[6];
    const float* b_dt   = (const float*)d_in[7];
    const float* A_log  = (const float*)d_in[8];
    const float* Dv     = (const float*)d_in[9];
    const float* W_out  = (const float*)d_in[10];
    float* out = (float*)d_out;

    // workspace carve-up (floats)
    float* ws = (float*)d_ws;
    size_t o = 0;
    float* xr   = ws + o; o += (size_t)BS * LSEQ * 2 * DI;   // 128 MB
    float* xc   = ws + o; o += (size_t)BS * LSEQ * DI;       //  64 MB
    float* proj = ws + o; o += (size_t)BS * LSEQ * 64;       //   4 MB
    float* dty  = ws + o; o += (size_t)BS * LSEQ * DI;       //  64 MB (dt, reused as y)
    float* cur0 = ws + o; o += (size_t)BS * LSEQ * DM;       //  32 MB

    for (int l = 0; l < NLAYER; ++l) {
        const float* inp  = (l == 0) ? x    : cur0;
        float*       outp = (l == 0) ? cur0 : out;           // last layer -> d_out

        // xr = inp @ W_in[e,l]            M=L, N=2*DI, K=DM
        gemm_wmma_f32<<<dim3(2 * DI / NT, LSEQ / MT, BS), 256, 0, stream>>>(
            inp, W_in, xr, eid, l, (long long)DM * 2 * DI,
            LSEQ, 2 * DI, DM, DM, 2 * DI, 2 * DI, nullptr, 0);

        // xc = silu(causal_conv(xr[:, :DI]) + b)
        conv_silu_kernel<<<(int)(((size_t)BS * LSEQ * DI) / 256), 256, 0, stream>>>(
            xr, conv_w, conv_b, eid, l, xc);

        // proj = xc @ W_x[e,l]            M=L, N=64, K=DI
        gemm_wmma_f32<<<dim3(1, LSEQ / MT, BS), 256, 0, stream>>>(
            xc, W_x, proj, eid, l, (long long)DI * 64,
            LSEQ, 64, DI, DI, 64, 64, nullptr, 0);

        // dt = softplus(proj[:, :32] @ W_dt[e,l] + b_dt[e,l])   M=L, N=DI, K=32
        gemm_wmma_f32<<<dim3(DI / NT, LSEQ / MT, BS), 256, 0, stream>>>(
            proj, W_dt, dty, eid, l, (long long)DTR * DI,
            LSEQ, DI, DTR, 64, DI, DI, b_dt, 1);

        // selective scan (+gating); y written in-place over dt buffer
        scan_kernel<<<dim3(DI / 256, BS), 256, 0, stream>>>(
            proj, dty, xc, xr, A_log, Dv, eid, l, dty);

        // layer_out = y @ W_out[e,l]      M=L, N=DM, K=DI
        gemm_wmma_f32<<<dim3(DM / NT, LSEQ / MT, BS), 256, 0, stream>>>(
            dty, W_out, outp, eid, l, (long long)DI * DM,
            LSEQ, DM, DI, DI, DM, DM, nullptr, 0);
    }

    // expert_id == 0 -> passthrough input
    final_select_kernel<<<(int)(((size_t)BS * LSEQ * DM) / 256), 256, 0, stream>>>(
        x, eid, out);
}